// GroupAttention_one_dim_68659347193900
// MI455X (gfx1250) — compile-verified
//
#include <hip/hip_runtime.h>
#include <math.h>

// Problem constants (from setup_inputs): B=4, S=256, D=512
#define BB 4
#define SS 256
#define DD 512

typedef float v2f __attribute__((ext_vector_type(2)));
typedef float v8f __attribute__((ext_vector_type(8)));

// ---------------------------------------------------------------------------
// Kernel 1: per-row layernorm stats + dual dot products via FP32 WMMA.
// Grid: (B*S)/16 = 64 blocks, block = 32 threads (one wave32 per block).
// Each wave computes D = Xnorm(16 x 512) @ W(512 x 16), where W col0 = w1,
// col1 = w2, rest zero. Accumulated with V_WMMA_F32_16X16X4_F32 over 128
// K-chunks of 4. Output column 0 -> a[row], column 1 -> c[row].
// B-operand is built branch-free: per-lane weight pointer + 0/1 scale,
// so the hot loop is loads -> packed VALU -> v_wmma with no EXEC juggling.
// ---------------------------------------------------------------------------
__global__ __launch_bounds__(32)
void rowdots_wmma(const float* __restrict__ X,
                  const float* __restrict__ gamma,
                  const float* __restrict__ beta,
                  const float* __restrict__ wfc,
                  float* __restrict__ aout,
                  float* __restrict__ cout)
{
    const int lane    = threadIdx.x;          // 0..31
    const int rowbase = blockIdx.x * 16;      // 16 rows per wave
    const int n       = lane & 15;            // N-column role in WMMA B/D
    const int hl      = lane >> 4;            // half-wave id
    const int kofs    = hl << 1;              // K offset for A/B per ISA layout

    // ---- per-row mean / (std+eps)^-1, single pass, wave reductions ----
    float my_m = 0.f, my_inv = 0.f;
    for (int r = 0; r < 16; ++r) {
        const float* xr = X + (size_t)(rowbase + r) * DD;
        float s = 0.f, s2 = 0.f;
        for (int t = lane; t < DD; t += 32) {
            float v = xr[t];
            s += v; s2 += v * v;
        }
        for (int o = 16; o > 0; o >>= 1) {
            s  += __shfl_xor(s,  o, 32);
            s2 += __shfl_xor(s2, o, 32);
        }
        float m   = s * (1.0f / DD);
        float var = (s2 - (float)DD * m * m) * (1.0f / (DD - 1));
        float sd  = sqrtf(fmaxf(var, 0.0f));
        float inv = 1.0f / (sd + 1e-6f);
        if (n == r) { my_m = m; my_inv = inv; }   // lane keeps stats of its A-row
    }

    const float* w1 = wfc;
    const float* w2 = wfc + DD;
    const float* xrow = X + (size_t)(rowbase + n) * DD;

    // Branch-free B operand: column 0 reads w1, column 1 reads w2,
    // columns 2..15 read w1 but scale to zero. No predicated loads.
    const float* wp     = (n == 1) ? w2 : w1;
    const float  wscale = (n < 2) ? 1.0f : 0.0f;

    v8f acc = {0.f, 0.f, 0.f, 0.f, 0.f, 0.f, 0.f, 0.f};

    // ---- K loop: A = layernormed 16x4 tile, B = [w1 w2 0...] 4x16 tile ----
#pragma unroll 4
    for (int kk = 0; kk < DD; kk += 4) {
        const int d = kk + kofs;
        float2 xv = *(const float2*)(xrow  + d);
        float2 gv = *(const float2*)(gamma + d);
        float2 bv = *(const float2*)(beta  + d);
        float2 wv = *(const float2*)(wp    + d);

        v2f A;
        A.x = gv.x * (xv.x - my_m) * my_inv + bv.x;
        A.y = gv.y * (xv.y - my_m) * my_inv + bv.y;

        v2f Bv;
        Bv.x = wv.x * wscale;
        Bv.y = wv.y * wscale;

        acc = __builtin_amdgcn_wmma_f32_16x16x4_f32(
            /*neg_a=*/false, A, /*neg_b=*/false, Bv,
            /*c_mod=*/(short)0, acc, /*reuse_a=*/false, /*reuse_b=*/false);
    }

    // ---- D layout: lane holds N = n, VGPR v holds M = v + hl*8 ----
    if (n == 0) {
        for (int v = 0; v < 8; ++v) aout[rowbase + hl * 8 + v] = acc[v];
    } else if (n == 1) {
        for (int v = 0; v < 8; ++v) cout[rowbase + hl * 8 + v] = acc[v];
    }
}

// ---------------------------------------------------------------------------
// Kernel 2: per (b,i) row: prefix-sum of c -> P, logits, sigmoid, prior blend,
// Mij store, inclusive product scan (cumprod), symmetric G store, diag = 1.
// Grid: B*S = 1024 blocks, block = S = 256 threads (thread = column j).
// ---------------------------------------------------------------------------
__global__ __launch_bounds__(SS)
void fuse_kernel(const float* __restrict__ prior,
                 const float* __restrict__ bfcp,
                 const float* __restrict__ aws,
                 const float* __restrict__ cws,
                 float* __restrict__ Gout,
                 float* __restrict__ Mout)
{
    __shared__ float sc[SS];
    __shared__ float Ps[SS];

    const int b = blockIdx.x / SS;
    const int i = blockIdx.x % SS;
    const int j = threadIdx.x;

    // inclusive sum scan of c[b,:] -> P
    sc[j] = cws[b * SS + j];
    __syncthreads();
    for (int off = 1; off < SS; off <<= 1) {
        float t = (j >= off) ? sc[j - off] : 0.0f;
        __syncthreads();
        sc[j] += t;
        __syncthreads();
    }
    Ps[j] = sc[j];
    __syncthreads();

    const float Pj  = Ps[j];
    const float Qi  = (i > 0) ? Ps[i - 1] : 0.0f;
    const float aj  = aws[b * SS + j];
    const float bfc = bfcp[0];

    const bool valid = (j >= i);
    float logits = bfc;
    if (valid) logits += aj + (Pj - Qi) / (float)(j - i + 1);

    const float sg = 1.0f / (1.0f + __expf(-logits));
    const float p  = prior[((size_t)(b * SS + i)) * SS + j];
    const float M  = p + (1.0f - p) * sg;
    Mout[((size_t)(b * SS + i)) * SS + j] = M;

    // inclusive product scan (cumprod along j), identity 1 for j < i
    sc[j] = valid ? M : 1.0f;
    __syncthreads();
    for (int off = 1; off < SS; off <<= 1) {
        float t = (j >= off) ? sc[j - off] : 1.0f;
        __syncthreads();
        sc[j] *= t;
        __syncthreads();
    }
    const float g = sc[j];

    if (j == i) {
        Gout[((size_t)(b * SS + i)) * SS + i] = 1.0f;
    } else if (j > i) {
        Gout[((size_t)(b * SS + i)) * SS + j] = g;   // upper triangle
        Gout[((size_t)(b * SS + j)) * SS + i] = g;   // mirrored lower triangle
    }
}

extern "C" void kernel_launch(void* const* d_in, const int* in_sizes, int n_in,
                              void* d_out, int out_size, void* d_ws, size_t ws_size,
                              hipStream_t stream)
{
    const float* ctx   = (const float*)d_in[0];  // (B,S,D) f32
    const float* prior = (const float*)d_in[1];  // (B,S,S) f32
    // d_in[2] = eos_mask (unused by reference)
    const float* gamma = (const float*)d_in[3];  // (D,)
    const float* beta  = (const float*)d_in[4];  // (D,)
    const float* wfc   = (const float*)d_in[5];  // (2D,)
    const float* bfc   = (const float*)d_in[6];  // scalar

    float* Gout = (float*)d_out;                    // (B,S,S)
    float* Mout = Gout + (size_t)BB * SS * SS;      // (B,S,S)

    float* aws = (float*)d_ws;        // B*S floats
    float* cws = aws + BB * SS;       // B*S floats

    rowdots_wmma<<<(BB * SS) / 16, 32, 0, stream>>>(ctx, gamma, beta, wfc, aws, cws);
    fuse_kernel<<<BB * SS, SS, 0, stream>>>(prior, bfc, aws, cws, Gout, Mout);
}